// PairwiseDistance_42992622633413
// MI455X (gfx1250) — compile-verified
//
#include <hip/hip_runtime.h>

typedef float v2f __attribute__((ext_vector_type(2)));
typedef float v4f __attribute__((ext_vector_type(4)));
typedef float v8f __attribute__((ext_vector_type(8)));

// One wave32 computes one 16x16 tile of out[b] = ||x_i||^2 + ||y_j||^2 - 2 x.y^T
// using V_WMMA_F32_16X16X4_F32 over K = d = 64 (16 WMMA steps, f32-exact).
__global__ __launch_bounds__(256) void PairwiseDistance_wmma_f32(
    const float* __restrict__ x,   // [B, n, 64]
    const float* __restrict__ y,   // [B, n, 64]
    float* __restrict__ out,       // [B, n, n]
    int n, int tiles_per_dim) {
  const int lane = threadIdx.x & 31;
  const int wave = threadIdx.x >> 5;

  const int tile = blockIdx.x * 8 + wave;
  const int tiles_per_batch = tiles_per_dim * tiles_per_dim;
  const int b  = tile / tiles_per_batch;
  const int r  = tile - b * tiles_per_batch;
  const int ti = r / tiles_per_dim;          // 16-row tile index (i / M dim)
  const int tj = r - ti * tiles_per_dim;     // 16-col tile index (j / N dim)

  const int d = 64;
  const float* __restrict__ xb = x + (size_t)b * n * d;
  const float* __restrict__ yb = y + (size_t)b * n * d;

  // WMMA f32 16x16x4 operand layout (wave32):
  //   A (16x4, MxK): lane L holds row M=L&15; VGPR pair = K = {2*(L>>4), 2*(L>>4)+1}
  //   B (4x16, KxN): lane L holds col N=L&15; VGPR pair = K = {2*(L>>4), 2*(L>>4)+1}
  const int mn   = lane & 15;
  const int koff = (lane >> 4) * 2;

  const float* __restrict__ arow = xb + (size_t)(ti * 16 + mn) * d;  // x row for A
  const float* __restrict__ brow = yb + (size_t)(tj * 16 + mn) * d;  // y row for B (col of y^T)
  const float* __restrict__ aptr = arow + koff;
  const float* __restrict__ bptr = brow + koff;

  v8f acc = {};
#pragma unroll
  for (int k = 0; k < d; k += 4) {
    v2f av = *(const v2f*)(aptr + k);
    v2f bv = *(const v2f*)(bptr + k);
    // 8 args: (neg_a, A, neg_b, B, c_mod, C, reuse_a, reuse_b)
    acc = __builtin_amdgcn_wmma_f32_16x16x4_f32(false, av, false, bv,
                                                (short)0, acc, false, false);
  }

  // Row/col squared norms: lane computes ||x[ti*16+mn]||^2 and ||y[tj*16+mn]||^2.
  float xn = 0.f, yn = 0.f;
#pragma unroll
  for (int k = 0; k < d; k += 4) {
    v4f xv = *(const v4f*)(arow + k);
    v4f yv = *(const v4f*)(brow + k);
    xn = fmaf(xv.x, xv.x, xn); xn = fmaf(xv.y, xv.y, xn);
    xn = fmaf(xv.z, xv.z, xn); xn = fmaf(xv.w, xv.w, xn);
    yn = fmaf(yv.x, yv.x, yn); yn = fmaf(yv.y, yv.y, yn);
    yn = fmaf(yv.z, yv.z, yn); yn = fmaf(yv.w, yv.w, yn);
  }

  // C/D layout: VGPR v of lane L holds element (M = v + 8*(L>>4), N = L&15).
  // yn is already this lane's column norm. xn for row M lives in lane M:
  // broadcast with ds_bpermute (byte index = srcLane*4).
  const int mbase = (lane >> 4) * 8;
  const int xn_bits = __builtin_bit_cast(int, xn);

  float* __restrict__ op =
      out + (size_t)b * n * n + (size_t)(ti * 16) * n + (size_t)(tj * 16) + mn;

  float res[8];
#pragma unroll
  for (int v = 0; v < 8; ++v) {
    const int src = mbase + v;  // source lane holding xnorm of row M
    const float xnm = __builtin_bit_cast(
        float, __builtin_amdgcn_ds_bpermute(src << 2, xn_bits));
    res[v] = fmaf(-2.0f, acc[v], xnm + yn);
  }
#pragma unroll
  for (int v = 0; v < 8; ++v) {
    op[(size_t)(mbase + v) * n] = res[v];
  }
}

extern "C" void kernel_launch(void* const* d_in, const int* in_sizes, int n_in,
                              void* d_out, int out_size, void* d_ws, size_t ws_size,
                              hipStream_t stream) {
  (void)d_ws; (void)ws_size; (void)out_size;
  const float* x = (const float*)d_in[0];
  const float* y = (const float*)d_in[1];
  float* out = (float*)d_out;

  const int B = 2, d = 64;
  int n = 2048;
  if (n_in >= 1 && in_sizes && in_sizes[0] > 0) n = in_sizes[0] / (B * d);

  const int tpd = n / 16;                    // 16x16 tiles per dimension
  const int total_tiles = B * tpd * tpd;     // one wave per tile
  const int blocks = total_tiles / 8;        // 8 waves (256 threads) per block

  PairwiseDistance_wmma_f32<<<blocks, 256, 0, stream>>>(x, y, out, n, tpd);
}